// FinePredictor_11106785427922
// MI455X (gfx1250) — compile-verified
//
#include <hip/hip_runtime.h>
#include <hip/hip_bf16.h>

typedef __attribute__((ext_vector_type(2))) float v2f;
typedef __attribute__((ext_vector_type(8))) float v8f;

constexpr int Bn = 32;     // batch
constexpr int T  = 1024;   // seq len
constexpr int D  = 1024;   // hidden dim
constexpr int K  = 8;      // fine slots
constexpr int E  = 400;    // label emb dim

constexpr int TILE_T = 128;        // tokens per block in tokscore kernel
constexpr int CH     = 32;         // d-chunk staged in LDS
constexpr int HSTR   = CH + 4;     // padded LDS row stride (bank-conflict free: 4m+{0..3} covers 64 banks)

// -------------------------------------------------------------------------
// Kernel 1: tempT[b][d][k] = sum_e slot[b][e][k] * W[e][d]   (K-major for fast B-fragment reads)
// -------------------------------------------------------------------------
__global__ __launch_bounds__(256) void temp_kernel(const float* __restrict__ slot,
                                                   const float* __restrict__ W,
                                                   float* __restrict__ tempT) {
    const int b = blockIdx.y;
    const int d = blockIdx.x * 256 + threadIdx.x;
    __shared__ float s[E * K];                       // 12.8 KB
    for (int i = threadIdx.x; i < E * K; i += 256)
        s[i] = slot[(size_t)b * E * K + i];
    __syncthreads();
    float acc[K] = {};
    for (int e = 0; e < E; ++e) {
        float w = W[(size_t)e * D + d];              // coalesced; W resident in 192MB L2
        #pragma unroll
        for (int k = 0; k < K; ++k) acc[k] = fmaf(s[e * K + k], w, acc[k]);
    }
    #pragma unroll
    for (int k = 0; k < K; ++k)
        tempT[((size_t)b * D + d) * K + k] = acc[k];
}

// -------------------------------------------------------------------------
// Kernel 2: tokscore[b][t][k] = hidden[b][t][:] . tempT[b][:][k]  via V_WMMA_F32_16X16X4_F32
//   Bandwidth pass (128 MB of hidden). Double-buffered LDS staging with
//   GLOBAL_LOAD_ASYNC_TO_LDS_B128 (ASYNCcnt) so HBM streaming overlaps WMMA.
//   block = 256 threads (8 waves) = one batch x 128 tokens; each wave owns a
//   16x16 WMMA tile (N: 8 real slots + 8 zero-padded columns).
// -------------------------------------------------------------------------
__global__ __launch_bounds__(256) void tokscore_kernel(const float* __restrict__ hidden,
                                                       const float* __restrict__ tempT,
                                                       float* __restrict__ tok) {
    const int b     = blockIdx.y;
    const int tbase = blockIdx.x * TILE_T;
    const int tid   = threadIdx.x;
    const int wave  = tid >> 5;
    const int lane  = tid & 31;
    const int mn    = lane & 15;            // M row (A) / N col (B) within tile
    const int kk    = (lane >> 4) << 1;     // 0 for lanes 0-15, 2 for lanes 16-31 (ISA 16x4 f32 layout)

    __shared__ float h_tile[2][TILE_T * HSTR]; // 2 x 18 KB, padded stride -> conflict-free A reads
    __shared__ float t_lds[D * K];             // 32 KB: whole tempT[b]

    const float* hb = hidden + ((size_t)b * T + tbase) * D;
    const float* tp = tempT + (size_t)b * D * K;
    for (int i = tid; i < D * K; i += 256) t_lds[i] = tp[i];

    // Issue one chunk's worth of async global->LDS copies (4 b128 per thread,
    // 128B contiguous per row => fully coalesced; no VGPR data movement).
    auto issue_chunk = [&](int buf, int d0) {
        #pragma unroll
        for (int it = 0; it < (TILE_T * (CH / 4)) / 256; ++it) {
            int q = tid + it * 256;
            int r = q >> 3;                 // CH/4 == 8
            int j = (q & 7) << 2;
            unsigned long long ga = (unsigned long long)(uintptr_t)(hb + (size_t)r * D + d0 + j);
            unsigned int       la = (unsigned int)(uintptr_t)(&h_tile[buf][r * HSTR + j]);
            asm volatile("global_load_async_to_lds_b128 %0, %1, off"
                         :: "v"(la), "v"(ga) : "memory");
        }
    };

    issue_chunk(0, 0);                      // prologue: chunk 0 -> buffer 0

    v8f c = {0.f, 0.f, 0.f, 0.f, 0.f, 0.f, 0.f, 0.f};
    const int wrow = wave * 16 + mn;        // A-matrix row in tile for this lane

    for (int d0 = 0, itc = 0; d0 < D; d0 += CH, ++itc) {
        const int buf = itc & 1;
        const bool more = (d0 + CH) < D;
        if (more) issue_chunk(buf ^ 1, d0 + CH);   // keep next chunk in flight
        // Async loads complete in order: allowing 4 outstanding means this
        // wave's current-chunk copies have landed; next chunk still streams.
        if (more) asm volatile("s_wait_asynccnt 0x4" ::: "memory");
        else      asm volatile("s_wait_asynccnt 0x0" ::: "memory");
        __syncthreads();                    // cross-wave visibility of buf (also covers t_lds preload)

        #pragma unroll
        for (int dd = 0; dd < CH; dd += 4) {
            // A fragment: 16x4 f32, v0 = K{0|2}, v1 = K{1|3} per half-wave
            v2f a;
            a.x = h_tile[buf][wrow * HSTR + dd + kk];
            a.y = h_tile[buf][wrow * HSTR + dd + kk + 1];
            // B fragment: 4x16 f32, row kk striped over lanes; cols >= K zero-padded
            int dg = d0 + dd + kk;
            float bx = t_lds[dg * K + (mn & 7)];
            float by = t_lds[(dg + 1) * K + (mn & 7)];
            v2f bf;
            bf.x = (mn < K) ? bx : 0.f;     // select, not branch: EXEC stays all-ones for WMMA
            bf.y = (mn < K) ? by : 0.f;
            c = __builtin_amdgcn_wmma_f32_16x16x4_f32(false, a, false, bf, (short)0, c,
                                                      false, false);
        }
        __syncthreads();                    // all waves done reading buf before it is re-issued
    }

    // C/D layout: VGPR r -> M = r (lanes 0-15) or 8+r (lanes 16-31); N = lane&15
    float* ob = tok + ((size_t)b * T + tbase + wave * 16) * K;
    if (mn < K) {
        const int Mb = (lane < 16) ? 0 : 8;
        #pragma unroll
        for (int r = 0; r < 8; ++r) ob[(size_t)(Mb + r) * K + mn] = c[r];
    }
}

// -------------------------------------------------------------------------
// Kernel 3: deterministic span segment-sum + mask. One block per batch.
//   Spans are contiguous token ranges [start of s-th B-tag, start of next B-tag)
//   filtered to tokens labeled B or I. Invalid span slots output zero (mask +
//   output initialization; no atomics -> bitwise deterministic).
// -------------------------------------------------------------------------
__global__ __launch_bounds__(256) void span_kernel(const float* __restrict__ tok,
                                                   const int* __restrict__ labels,
                                                   const int* __restrict__ pB,
                                                   const int* __restrict__ pI,
                                                   float* __restrict__ out) {
    const int b   = blockIdx.x;
    const int tid = threadIdx.x;
    const int Bi  = *pB;
    const int Ii  = *pI;

    __shared__ int labs[T];
    __shared__ int cnt[256];
    __shared__ int starts[T + 1];
    __shared__ int nB;

    const int* lb = labels + (size_t)b * T;
    const int base = tid * 4;
    int cc = 0;
    #pragma unroll
    for (int j = 0; j < 4; ++j) {
        int l = lb[base + j];
        labs[base + j] = l;
        cc += (l == Bi) ? 1 : 0;
    }
    cnt[tid] = cc;
    __syncthreads();
    if (tid == 0) {                          // exclusive scan of 256 partials (cheap)
        int run = 0;
        for (int i = 0; i < 256; ++i) { int t = cnt[i]; cnt[i] = run; run += t; }
        nB = run;
        starts[run] = T;                     // sentinel end
    }
    __syncthreads();
    int run = cnt[tid];
    #pragma unroll
    for (int j = 0; j < 4; ++j) {
        if (labs[base + j] == Bi) { starts[run] = base + j; run++; }
    }
    __syncthreads();

    const float* tb = tok + (size_t)b * T * K;
    float*       ob = out + (size_t)b * T * K;
    const int ns = nB;
    for (int s = tid; s < T; s += 256) {
        float acc[K] = {};
        if (s < ns) {
            const int t0 = starts[s], t1 = starts[s + 1];
            for (int t = t0; t < t1; ++t) {
                int l = labs[t];
                if (l == Bi || l == Ii) {
                    #pragma unroll
                    for (int k = 0; k < K; ++k) acc[k] += tb[(size_t)t * K + k];
                }
            }
        }
        #pragma unroll
        for (int k = 0; k < K; ++k) ob[(size_t)s * K + k] = acc[k];
    }
}

// -------------------------------------------------------------------------
extern "C" void kernel_launch(void* const* d_in, const int* in_sizes, int n_in,
                              void* d_out, int out_size, void* d_ws, size_t ws_size,
                              hipStream_t stream) {
    const float* hidden = (const float*)d_in[0];   // [B,T,D] f32
    const float* slot   = (const float*)d_in[1];   // [B,E,K] f32
    const float* W      = (const float*)d_in[2];   // [E,D]   f32
    const int*   labels = (const int*)d_in[3];     // [B,T]   i32
    const int*   pB     = (const int*)d_in[4];     // scalar
    const int*   pI     = (const int*)d_in[5];     // scalar
    float*       out    = (float*)d_out;           // [B,T,K] f32

    float* tempT = (float*)d_ws;                         // [B,D,K] = 1 MB
    float* tok   = tempT + (size_t)Bn * D * K;           // [B,T,K] = 1 MB

    temp_kernel<<<dim3(D / 256, Bn), 256, 0, stream>>>(slot, W, tempT);
    tokscore_kernel<<<dim3(T / TILE_T, Bn), 256, 0, stream>>>(hidden, tempT, tok);
    span_kernel<<<dim3(Bn), 256, 0, stream>>>(tok, labels, pB, pI, out);
}